// GraphAttentionLayer_30193620091224
// MI455X (gfx1250) — compile-verified
//
#include <hip/hip_runtime.h>
#include <math.h>

#define NN   4096
#define EE   131072
#define DD   256
#define HH   4
#define HIDD 64
#define HD   256      // H * HID
#define SLOPE 0.2f
#define MAXDEG 256

typedef __attribute__((ext_vector_type(2))) float v2f;
typedef __attribute__((ext_vector_type(8))) float v8f;

// ---------------------------------------------------------------------------
// Kernel 1: Wh = input(4096x256) @ W(256x256) using V_WMMA_F32_16X16X4_F32.
// One wave per 16x16 output tile; 8 waves per block.
// A layout (16x4 f32): lane = M (0..15, repeated), VGPR0 = K=0 (lanes 0-15) /
//   K=2 (lanes 16-31), VGPR1 = K=1 / K=3.
// B layout (4x16 f32): lane = N, VGPR0 = K=0 / K=2, VGPR1 = K=1 / K=3.
// C/D (16x16 f32): VGPR r -> M = r (lanes 0-15), M = r+8 (lanes 16-31), N = lane&15.
// ---------------------------------------------------------------------------
__global__ __launch_bounds__(256) void gat_gemm(const float* __restrict__ A,
                                                const float* __restrict__ B,
                                                float* __restrict__ C) {
  const int wave = threadIdx.x >> 5;
  const int lane = threadIdx.x & 31;
  const int tile = blockIdx.x * 8 + wave;   // 0..4095
  const int tm = tile >> 4;                 // 0..255 row tile
  const int tn = tile & 15;                 // 0..15  col tile
  const int half = lane >> 4;               // 0 or 1
  const int mn = lane & 15;

  const float* Ar = A + (tm * 16 + mn) * DD;   // this lane's A row
  const float* Bc = B + tn * 16 + mn;          // this lane's B column

  v8f c = {};
  for (int kb = 0; kb < DD; kb += 4) {
    const int k0 = kb + 2 * half;
    v2f a, b;
    a.x = Ar[k0];
    a.y = Ar[k0 + 1];
    b.x = Bc[(size_t)k0 * HD];
    b.y = Bc[(size_t)(k0 + 1) * HD];
    c = __builtin_amdgcn_wmma_f32_16x16x4_f32(
        /*neg_a=*/false, a, /*neg_b=*/false, b,
        /*c_mod=*/(short)0, c, /*reuse_a=*/false, /*reuse_b=*/false);
  }

  float* Cr = C + (size_t)(tm * 16 + 8 * half) * HD + tn * 16 + mn;
#pragma unroll
  for (int r = 0; r < 8; ++r) Cr[(size_t)r * HD] = c[r];
}

// ---------------------------------------------------------------------------
// Kernel 2: e1[i,h] = dot(Wh[i, h*64 : h*64+64], a1)   (a1 = attention[:64])
//           e2[i,h] = dot(Wh[i, h*64 : h*64+64], a2)   (a2 = attention[64:])
// ---------------------------------------------------------------------------
__global__ void gat_e(const float* __restrict__ Wh, const float* __restrict__ att,
                      float* __restrict__ e1, float* __restrict__ e2) {
  const int t = blockIdx.x * blockDim.x + threadIdx.x;   // N*H threads
  if (t >= NN * HH) return;
  const int i = t >> 2;
  const int h = t & 3;
  const float* w = Wh + (size_t)i * HD + h * HIDD;
  float s1 = 0.f, s2 = 0.f;
#pragma unroll 8
  for (int f = 0; f < HIDD; ++f) {
    const float v = w[f];
    s1 += v * att[f];
    s2 += v * att[HIDD + f];
  }
  e1[t] = s1;
  e2[t] = s2;
}

// ---------------------------------------------------------------------------
// Kernel 3: adjacency bitmask init: zero + diagonal (self loop), dedups edges
// exactly like the reference's dense binary adjacency.
// ---------------------------------------------------------------------------
__global__ void gat_adj_init(unsigned int* __restrict__ mask) {
  const int t = blockIdx.x * blockDim.x + threadIdx.x;   // N*128 words
  const int row = t >> 7;
  const int w = t & 127;
  mask[t] = ((row >> 5) == w) ? (1u << (row & 31)) : 0u;
}

// Kernel 4: scatter edges into bitmask rows (row = node_in, bit = node_out)
__global__ void gat_adj_edges(const int* __restrict__ el,
                              unsigned int* __restrict__ mask) {
  const int e = blockIdx.x * blockDim.x + threadIdx.x;
  if (e >= EE) return;
  const int i = el[e * 3 + 0];
  const int j = el[e * 3 + 1];
  atomicOr(&mask[(size_t)i * 128 + (j >> 5)], 1u << (j & 31));
}

// ---------------------------------------------------------------------------
// Kernel 5: one wave32 per node. Extract neighbor list from the bit row into
// LDS, then per-head softmax (2-pass: max, then sum + weighted accumulation
// of Wh[j]) and ELU. Lane L owns output features [L*8, L*8+8) -> head = L>>3.
// ---------------------------------------------------------------------------
__global__ __launch_bounds__(256) void gat_aggregate(
    const float* __restrict__ Wh, const float* __restrict__ e1,
    const float* __restrict__ e2, const unsigned int* __restrict__ mask,
    float* __restrict__ out) {
  __shared__ int s_nbr[8][MAXDEG];
  __shared__ int s_cnt[8];

  const int wave = threadIdx.x >> 5;
  const int lane = threadIdx.x & 31;
  const int i = blockIdx.x * 8 + wave;

  if (lane == 0) s_cnt[wave] = 0;
  __syncthreads();

  // Extract set bits of row i (128 words) into the wave's neighbor list.
#pragma unroll
  for (int w4 = 0; w4 < 4; ++w4) {
    const int word = w4 * 32 + lane;
    unsigned int bits = mask[(size_t)i * 128 + word];
    while (bits) {
      const int b = __builtin_ctz(bits);
      bits &= bits - 1u;
      const int pos = atomicAdd(&s_cnt[wave], 1);
      if (pos < MAXDEG) s_nbr[wave][pos] = word * 32 + b;
    }
  }
  __syncthreads();

  int deg = s_cnt[wave];
  if (deg > MAXDEG) deg = MAXDEG;   // degree ~33 avg; cap is pure safety

  const int h = lane >> 3;           // head owned by this lane
  const int fbase = lane * 8;        // 8 contiguous features, all in head h
  const float e1i = e1[i * HH + h];

  // Pass 1: row max of leakyrelu(e1[i,h] + e2[j,h])
  float m = -INFINITY;
  for (int t = 0; t < deg; ++t) {
    const int j = s_nbr[wave][t];
    const float v = e1i + e2[j * HH + h];
    const float lr = (v >= 0.f) ? v : SLOPE * v;
    m = fmaxf(m, lr);
  }

  // Pass 2: exp-sum + weighted feature accumulation
  float sum = 0.f;
  float acc[8] = {0.f, 0.f, 0.f, 0.f, 0.f, 0.f, 0.f, 0.f};
  for (int t = 0; t < deg; ++t) {
    const int j = s_nbr[wave][t];
    const float v = e1i + e2[j * HH + h];
    const float lr = (v >= 0.f) ? v : SLOPE * v;
    const float p = __expf(lr - m);
    sum += p;
    const float4* wp = (const float4*)(Wh + (size_t)j * HD + fbase);
    const float4 w0 = wp[0];
    const float4 w1 = wp[1];
    acc[0] += p * w0.x; acc[1] += p * w0.y; acc[2] += p * w0.z; acc[3] += p * w0.w;
    acc[4] += p * w1.x; acc[5] += p * w1.y; acc[6] += p * w1.z; acc[7] += p * w1.w;
  }

  const float inv = 1.f / sum;       // deg >= 1 (self loop) -> sum > 0
#pragma unroll
  for (int k = 0; k < 8; ++k) {
    const float x = acc[k] * inv;
    out[(size_t)i * HD + fbase + k] = (x > 0.f) ? x : (__expf(x) - 1.f);
  }
}

// ---------------------------------------------------------------------------
extern "C" void kernel_launch(void* const* d_in, const int* in_sizes, int n_in,
                              void* d_out, int out_size, void* d_ws, size_t ws_size,
                              hipStream_t stream) {
  (void)in_sizes; (void)n_in; (void)out_size; (void)ws_size;
  const float* input     = (const float*)d_in[0];
  const int*   edge_list = (const int*)d_in[1];
  // d_in[2] = num_node scalar (N is compile-time here)
  const float* W         = (const float*)d_in[3];
  const float* att       = (const float*)d_in[4];
  float* out = (float*)d_out;

  // Workspace layout: Wh (4 MB) | e1 (64 KB) | e2 (64 KB) | adj bitmask (2 MB)
  float* Wh = (float*)d_ws;
  float* e1 = Wh + (size_t)NN * HD;
  float* e2 = e1 + (size_t)NN * HH;
  unsigned int* mask = (unsigned int*)(e2 + (size_t)NN * HH);

  gat_gemm<<<(NN / 16) * (HD / 16) / 8, 256, 0, stream>>>(input, W, Wh);
  gat_e<<<(NN * HH + 255) / 256, 256, 0, stream>>>(Wh, att, e1, e2);
  gat_adj_init<<<(NN * 128) / 256, 256, 0, stream>>>(mask);
  gat_adj_edges<<<EE / 256, 256, 0, stream>>>(edge_list, mask);
  gat_aggregate<<<NN / 8, 256, 0, stream>>>(Wh, e1, e2, mask, out);
}